// MaskedAutoEncoder_39256001086071
// MI455X (gfx1250) — compile-verified
//
#include <hip/hip_runtime.h>

typedef __attribute__((ext_vector_type(2))) float v2f;
typedef __attribute__((ext_vector_type(8))) float v8f;

#define BN 128
#define CCH 3
#define IMGD 256
#define EDIM 200
#define EPAD 208
#define PE 768
#define NVIS 64
#define NMASK 192
#define MROWS (BN * NVIS)                 // 8192
#define IMG_TOT (BN * CCH * IMGD * IMGD)  // 25165824

// ---------------------------------------------------------------------------
// GEMM1: emb[row, e] = sum_p patch(row, p) * conv_w[e, p] + conv_b[e] + pos
// rows = b*64+v over visible patches, A gathered straight from x.
// ---------------------------------------------------------------------------
__global__ __launch_bounds__(256) void mae_gemm1(
    const float* __restrict__ x, const float* __restrict__ conv_w,
    const float* __restrict__ conv_b, const float* __restrict__ pos_embed,
    const int* __restrict__ perm, float* __restrict__ emb) {
  __shared__ float Bs[16 * 772];  // [n-within-tile][k], stride 772 (4 mod 64)
  const int tid = threadIdx.x;
  const int n0 = blockIdx.x * 16;
  const int row0 = blockIdx.y * 128;

  for (int idx = tid; idx < PE * 16; idx += 256) {
    int p = idx >> 4, j = idx & 15;
    int e = n0 + j;
    Bs[j * 772 + p] = (e < EDIM) ? conv_w[e * PE + p] : 0.0f;
  }
  __syncthreads();

  const int wave = tid >> 5, lane = tid & 31;
  const int rloc = lane & 15;
  const int khalf = (lane >> 4) << 1;  // 0 or 2
  const int wrow0 = row0 + wave * 16;
  const int row = wrow0 + rloc;
  const int b = row >> 6;
  const int v = row & 63;
  const int patch = perm[NMASK + v];
  const int gy = patch >> 4, gx = patch & 15;
  const float* xb =
      x + (size_t)b * (CCH * IMGD * IMGD) + gy * (16 * IMGD) + gx * 16;
  const float* bsrow = Bs + rloc * 772;

  v8f acc = {};
  for (int k0 = 0; k0 < PE; k0 += 4) {
    int k = k0 + khalf;
    int c = k >> 8, pr = (k >> 4) & 15, pc = k & 15;
    v2f a = *(const v2f*)(xb + c * (IMGD * IMGD) + pr * IMGD + pc);
    v2f bb = *(const v2f*)(bsrow + k);
    acc = __builtin_amdgcn_wmma_f32_16x16x4_f32(false, a, false, bb, (short)0,
                                                acc, false, false);
  }

  const int col = n0 + rloc;
  const int rowbase = wrow0 + ((lane >> 4) << 3);
  const float cb = (col < EDIM) ? conv_b[col] : 0.0f;
#pragma unroll
  for (int r = 0; r < 8; ++r) {
    int rr = rowbase + r;
    int pat = perm[NMASK + (rr & 63)];
    float pe = (col < EDIM) ? pos_embed[pat * EDIM + col] : 0.0f;
    emb[(size_t)rr * EPAD + col] = acc[r] + cb + pe;
  }
}

// ---------------------------------------------------------------------------
// GEMM2: full[row, f] = emb[row, :] @ W_enc + b_enc[f] + pos_embed  (K=208 pad)
// ---------------------------------------------------------------------------
__global__ __launch_bounds__(256) void mae_gemm2(
    const float* __restrict__ emb, const float* __restrict__ W_enc,
    const float* __restrict__ b_enc, const float* __restrict__ pos_embed,
    const int* __restrict__ perm, float* __restrict__ full) {
  __shared__ float Bs[16 * 212];
  const int tid = threadIdx.x;
  const int n0 = blockIdx.x * 16;
  const int row0 = blockIdx.y * 128;

  for (int idx = tid; idx < EPAD * 16; idx += 256) {
    int k = idx >> 4, j = idx & 15;
    int e = n0 + j;
    Bs[j * 212 + k] = (e < EDIM && k < EDIM) ? W_enc[k * EDIM + e] : 0.0f;
  }
  __syncthreads();

  const int wave = tid >> 5, lane = tid & 31;
  const int rloc = lane & 15;
  const int khalf = (lane >> 4) << 1;
  const int wrow0 = row0 + wave * 16;
  const float* arow = emb + (size_t)(wrow0 + rloc) * EPAD;
  const float* bsrow = Bs + rloc * 212;

  v8f acc = {};
  for (int k0 = 0; k0 < EPAD; k0 += 4) {
    int k = k0 + khalf;
    v2f a = *(const v2f*)(arow + k);
    v2f bb = *(const v2f*)(bsrow + k);
    acc = __builtin_amdgcn_wmma_f32_16x16x4_f32(false, a, false, bb, (short)0,
                                                acc, false, false);
  }

  const int col = n0 + rloc;
  const int rowbase = wrow0 + ((lane >> 4) << 3);
  const float be = (col < EDIM) ? b_enc[col] : 0.0f;
#pragma unroll
  for (int r = 0; r < 8; ++r) {
    int rr = rowbase + r;
    int pat = perm[NMASK + (rr & 63)];
    float pe = (col < EDIM) ? pos_embed[pat * EDIM + col] : 0.0f;
    full[(size_t)rr * EPAD + col] = acc[r] + be + pe;
  }
}

// ---------------------------------------------------------------------------
// GEMM3: dec[row, p] = full[row, :] @ W_dec + b_dec[p], scattered (folded)
// straight into the output image at the visible patch positions.
// ---------------------------------------------------------------------------
__global__ __launch_bounds__(256) void mae_gemm3(
    const float* __restrict__ full, const float* __restrict__ W_dec,
    const float* __restrict__ b_dec, const int* __restrict__ perm,
    float* __restrict__ out) {
  __shared__ float Bs[16 * 212];
  const int tid = threadIdx.x;
  const int n0 = blockIdx.x * 16;  // p tile
  const int row0 = blockIdx.y * 128;

  for (int idx = tid; idx < EPAD * 16; idx += 256) {
    int k = idx >> 4, j = idx & 15;
    Bs[j * 212 + k] = (k < EDIM) ? W_dec[k * PE + n0 + j] : 0.0f;
  }
  __syncthreads();

  const int wave = tid >> 5, lane = tid & 31;
  const int rloc = lane & 15;
  const int khalf = (lane >> 4) << 1;
  const int wrow0 = row0 + wave * 16;
  const float* arow = full + (size_t)(wrow0 + rloc) * EPAD;
  const float* bsrow = Bs + rloc * 212;

  v8f acc = {};
  for (int k0 = 0; k0 < EPAD; k0 += 4) {
    int k = k0 + khalf;
    v2f a = *(const v2f*)(arow + k);
    v2f bb = *(const v2f*)(bsrow + k);
    acc = __builtin_amdgcn_wmma_f32_16x16x4_f32(false, a, false, bb, (short)0,
                                                acc, false, false);
  }

  const int p = n0 + rloc;
  const int c = p >> 8, pr = (p >> 4) & 15, pc = p & 15;
  const int rowbase = wrow0 + ((lane >> 4) << 3);
  const float bd = b_dec[p];
#pragma unroll
  for (int r = 0; r < 8; ++r) {
    int rr = rowbase + r;
    int bb = rr >> 6;
    int pat = perm[NMASK + (rr & 63)];
    int gy = pat >> 4, gx = pat & 15;
    size_t o = (size_t)bb * (CCH * IMGD * IMGD) + c * (IMGD * IMGD) +
               (gy * 16 + pr) * IMGD + gx * 16 + pc;
    out[o] = acc[r] + bd;
  }
}

// ---------------------------------------------------------------------------
// Masked-patch decode table: decm[j, p] = (mask_token + pos_embed[perm[j]]) @
// W_dec + b_dec. Batch-independent -> computed once. Also emits masked_idx.
// ---------------------------------------------------------------------------
__global__ __launch_bounds__(256) void mae_masked_dec(
    const float* __restrict__ mask_token, const float* __restrict__ pos_embed,
    const float* __restrict__ W_dec, const float* __restrict__ b_dec,
    const int* __restrict__ perm, float* __restrict__ decm,
    float* __restrict__ out) {
  __shared__ float rvec[EDIM];
  const int j = blockIdx.x;
  const int t = threadIdx.x;
  const int nj = perm[j];
  if (t < EDIM) rvec[t] = mask_token[t] + pos_embed[nj * EDIM + t];
  if (t == 0) ((int*)out)[IMG_TOT + j] = nj;  // masked_idx output (int32 bits)
  __syncthreads();
  for (int p = t; p < PE; p += 256) {
    float acc = b_dec[p];
    for (int e = 0; e < EDIM; ++e) acc += rvec[e] * W_dec[e * PE + p];
    decm[j * PE + p] = acc;
  }
}

// Broadcast the 192 precomputed masked-patch rows into every batch image.
__global__ __launch_bounds__(256) void mae_scatter_masked(
    const float* __restrict__ decm, const int* __restrict__ perm,
    float* __restrict__ out) {
  const int j = blockIdx.x % NMASK;
  const int b = blockIdx.x / NMASK;
  const int pat = perm[j];
  const int gy = pat >> 4, gx = pat & 15;
  const size_t ob = (size_t)b * (CCH * IMGD * IMGD);
  for (int p = threadIdx.x; p < PE; p += 256) {
    int c = p >> 8, pr = (p >> 4) & 15, pc = p & 15;
    out[ob + c * (IMGD * IMGD) + (gy * 16 + pr) * IMGD + gx * 16 + pc] =
        decm[j * PE + p];
  }
}

extern "C" void kernel_launch(void* const* d_in, const int* in_sizes, int n_in,
                              void* d_out, int out_size, void* d_ws,
                              size_t ws_size, hipStream_t stream) {
  const float* x = (const float*)d_in[0];
  const float* conv_w = (const float*)d_in[1];
  const float* conv_b = (const float*)d_in[2];
  const float* pos_embed = (const float*)d_in[3];
  const float* mask_token = (const float*)d_in[4];
  const float* W_enc = (const float*)d_in[5];
  const float* b_enc = (const float*)d_in[6];
  const float* W_dec = (const float*)d_in[7];
  const float* b_dec = (const float*)d_in[8];
  const int* perm = (const int*)d_in[9];
  float* out = (float*)d_out;

  float* emb = (float*)d_ws;                    // [8192][208]
  float* full = emb + (size_t)MROWS * EPAD;     // [8192][208]
  float* decm = full + (size_t)MROWS * EPAD;    // [192][768]

  mae_masked_dec<<<NMASK, 256, 0, stream>>>(mask_token, pos_embed, W_dec, b_dec,
                                            perm, decm, out);
  mae_gemm1<<<dim3(13, 64), 256, 0, stream>>>(x, conv_w, conv_b, pos_embed,
                                              perm, emb);
  mae_gemm2<<<dim3(13, 64), 256, 0, stream>>>(emb, W_enc, b_enc, pos_embed,
                                              perm, full);
  mae_gemm3<<<dim3(48, 64), 256, 0, stream>>>(full, W_dec, b_dec, perm, out);
  mae_scatter_masked<<<BN * NMASK, 256, 0, stream>>>(decm, perm, out);
}